// EncoderRNN_90340342104527
// MI455X (gfx1250) — compile-verified
//
#include <hip/hip_runtime.h>
#include <math.h>

typedef float v2f __attribute__((ext_vector_type(2)));
typedef float v8f __attribute__((ext_vector_type(8)));

#define HSZ    1000
#define KPAD   1024              // K padded to multiple of 32 (pad reads x=0)
#define LAY    4
#define GSZ    (4 * HSZ)         // 4000 gate rows per layer
#define GROUPS (GSZ / 16)        // 250 row-groups of 16
#define KS_GH  2                 // K-slices for recurrent GEMVs (slice = 512)
#define KS_GI  4                 // K-slices for input GEMVs    (slice = 256)
#define SL_GH  (KPAD / KS_GH)    // 512
#define SL_GI  (KPAD / KS_GI)    // 256

__device__ __forceinline__ v2f ldv2(const float* p) { return *(const v2f*)p; }

// ---------------------------------------------------------------------------
// Fully-specialized software-pipelined f32-WMMA dot-product slice.
// One wave computes 16 dot products (rows row0..row0+15) over K = [K0, K0+S).
// S and K0 are template constants -> every load is base + immediate offset,
// the pad clamp folds away, and the body is pure loads + wmma + waits.
// Double-buffered 8-pair blocks keep up to 32 global_load_b64 in flight.
// ---------------------------------------------------------------------------
template <int S, int K0>
__device__ __forceinline__ float dot_slice_wmma(const float* __restrict__ xb,
                                                const float* __restrict__ Wb) {
    constexpr int BLOCKS = S / 32;   // 8 or 16 (even)
    v8f accs[4] = {v8f{}, v8f{}, v8f{}, v8f{}};
    v2f aA[8], bA[8], aB[8], bB[8];

    auto loadblk = [&](v2f (&a)[8], v2f (&b)[8], int kb) {
#pragma unroll
        for (int u = 0; u < 8; ++u) {
            int k  = kb + 4 * u;            // constexpr after unroll
            int kc = (k >= 1000) ? 996 : k; // folds away at compile time
            a[u] = ldv2(xb + k);            // pad region reads zero
            b[u] = ldv2(Wb + kc);           // max elem = 996+2+1 = 999, in row
        }
    };
    auto macblk = [&](v2f (&a)[8], v2f (&b)[8]) {
#pragma unroll
        for (int u = 0; u < 8; ++u) {
            accs[u & 3] = __builtin_amdgcn_wmma_f32_16x16x4_f32(
                false, a[u], false, b[u], (short)0, accs[u & 3], false, false);
        }
    };

    loadblk(aA, bA, K0);
#pragma unroll
    for (int blk = 0; blk < BLOCKS; blk += 2) {
        loadblk(aB, bB, K0 + (blk + 1) * 32);
        macblk(aA, bA);
        if (blk + 2 < BLOCKS) loadblk(aA, bA, K0 + (blk + 2) * 32);
        macblk(aB, bB);
    }
    v8f t = accs[0] + accs[1] + accs[2] + accs[3];
    return t[0];   // D row M=0, lanes 0-15 hold the 16 results
}

// ---------------------------------------------------------------------------
// x = emb[idx], zero-padded to KPAD
// ---------------------------------------------------------------------------
__global__ void embed_pad_kernel(const int* __restrict__ idx,
                                 const float* __restrict__ emb,
                                 float* __restrict__ xpad) {
    int i = blockIdx.x * blockDim.x + threadIdx.x;
    if (i < KPAD) xpad[i] = (i < HSZ) ? emb[(size_t)idx[0] * HSZ + i] : 0.0f;
}

// hidden -> zero-padded copies (LAY x KPAD)
__global__ void hidden_pad_kernel(const float* __restrict__ hidden,
                                  float* __restrict__ hpad) {
    int i = blockIdx.x * blockDim.x + threadIdx.x;
    if (i >= LAY * KPAD) return;
    int l = i >> 10, t = i & (KPAD - 1);
    hpad[i] = (t < HSZ) ? hidden[l * HSZ + t] : 0.0f;
}

// ---------------------------------------------------------------------------
// Recurrent-path partial GEMVs, all layers x all K-slices in parallel:
//   ghp[l][sl][row] = dot(Whh[l][row, slice_sl], hidden[l][slice_sl])
// ---------------------------------------------------------------------------
__global__ void __launch_bounds__(32)
gh_kernel(const float* __restrict__ Whh, const float* __restrict__ hpad,
          float* __restrict__ ghp) {
    const int bid  = blockIdx.x;
    const int sl   = bid % KS_GH;
    const int grp  = (bid / KS_GH) % GROUPS;
    const int l    = bid / (KS_GH * GROUPS);
    const int lane = threadIdx.x;
    const int khalf = (lane >> 4) << 1;
    const int row   = grp * 16 + (lane & 15);

    const float* Wb = Whh + (size_t)l * GSZ * HSZ + (size_t)row * HSZ + khalf;
    const float* xb = hpad + l * KPAD + khalf;
    float r = (sl == 0) ? dot_slice_wmma<SL_GH, 0>(xb, Wb)
                        : dot_slice_wmma<SL_GH, SL_GH>(xb, Wb);
    if (lane < 16)
        ghp[((size_t)l * KS_GH + sl) * GSZ + row] = r;
}

// ---------------------------------------------------------------------------
// Input-path partial GEMVs for one layer (sequential on x):
//   gip[sl][row] = dot(Wih_l[row, slice_sl], x[slice_sl])
// ---------------------------------------------------------------------------
__global__ void __launch_bounds__(32)
gi_kernel(const float* __restrict__ Wih_l, const float* __restrict__ xpad,
          float* __restrict__ gip) {
    const int sl   = blockIdx.x % KS_GI;
    const int grp  = blockIdx.x / KS_GI;
    const int lane = threadIdx.x;
    const int khalf = (lane >> 4) << 1;
    const int row   = grp * 16 + (lane & 15);

    const float* Wb = Wih_l + (size_t)row * HSZ + khalf;
    const float* xb = xpad + khalf;
    float r;
    if      (sl == 0) r = dot_slice_wmma<SL_GI, 0 * SL_GI>(xb, Wb);
    else if (sl == 1) r = dot_slice_wmma<SL_GI, 1 * SL_GI>(xb, Wb);
    else if (sl == 2) r = dot_slice_wmma<SL_GI, 2 * SL_GI>(xb, Wb);
    else              r = dot_slice_wmma<SL_GI, 3 * SL_GI>(xb, Wb);
    if (lane < 16)
        gip[(size_t)sl * GSZ + row] = r;
}

// ---------------------------------------------------------------------------
// Reduce partials + biases, LSTM cell elementwise; h2 feeds next layer's x.
// ---------------------------------------------------------------------------
__global__ void cell_kernel(const float* __restrict__ ghp_l,
                            const float* __restrict__ gip,
                            const float* __restrict__ bih_l,
                            const float* __restrict__ bhh_l,
                            const float* __restrict__ Cprev,
                            float* __restrict__ xpad,
                            float* __restrict__ out_h,
                            float* __restrict__ out_c,
                            float* __restrict__ out_final,
                            int is_last) {
    int i = blockIdx.x * blockDim.x + threadIdx.x;
    if (i >= HSZ) return;

    float g[4];
#pragma unroll
    for (int j = 0; j < 4; ++j) {
        int r = j * HSZ + i;
        float s = bih_l[r] + bhh_l[r];
#pragma unroll
        for (int sl = 0; sl < KS_GH; ++sl) s += ghp_l[sl * GSZ + r];
#pragma unroll
        for (int sl = 0; sl < KS_GI; ++sl) s += gip[sl * GSZ + r];
        g[j] = s;
    }
    float si = 1.0f / (1.0f + __expf(-g[0]));
    float sf = 1.0f / (1.0f + __expf(-g[1]));
    float tg = tanhf(g[2]);
    float so = 1.0f / (1.0f + __expf(-g[3]));
    float c2 = sf * Cprev[i] + si * tg;
    float h2 = so * tanhf(c2);
    xpad[i]  = h2;   // pad region untouched, stays zero
    out_h[i] = h2;
    out_c[i] = c2;
    if (is_last) out_final[i] = h2;
}

// ---------------------------------------------------------------------------
// Orchestration. d_out layout (f32): [output 1000 | hs 4*1000 | cs 4*1000]
// ---------------------------------------------------------------------------
extern "C" void kernel_launch(void* const* d_in, const int* in_sizes, int n_in,
                              void* d_out, int out_size, void* d_ws, size_t ws_size,
                              hipStream_t stream) {
    const int*   inp    = (const int*)d_in[0];
    const float* hidden = (const float*)d_in[1];
    const float* C      = (const float*)d_in[2];
    const float* emb    = (const float*)d_in[3];
    const float* Wih    = (const float*)d_in[4];
    const float* Whh    = (const float*)d_in[5];
    const float* bih    = (const float*)d_in[6];
    const float* bhh    = (const float*)d_in[7];
    float* out = (float*)d_out;
    float* ws  = (float*)d_ws;

    float* xpad = ws;                       // KPAD
    float* hpad = ws + KPAD;                // LAY*KPAD
    float* gip  = ws + KPAD + LAY * KPAD;   // KS_GI*GSZ
    float* ghp  = gip + KS_GI * GSZ;        // LAY*KS_GH*GSZ

    embed_pad_kernel<<<(KPAD + 255) / 256, 256, 0, stream>>>(inp, emb, xpad);
    hidden_pad_kernel<<<(LAY * KPAD + 255) / 256, 256, 0, stream>>>(hidden, hpad);

    // All recurrent partial GEMVs in parallel: 2000 waves streaming 64 MB
    gh_kernel<<<LAY * GROUPS * KS_GH, 32, 0, stream>>>(Whh, hpad, ghp);

    // Sequential layer chain: 1000 waves streaming 16 MB per step
    for (int l = 0; l < LAY; ++l) {
        gi_kernel<<<GROUPS * KS_GI, 32, 0, stream>>>(
            Wih + (size_t)l * GSZ * HSZ, xpad, gip);
        cell_kernel<<<(HSZ + 255) / 256, 256, 0, stream>>>(
            ghp + (size_t)l * KS_GH * GSZ, gip,
            bih + (size_t)l * GSZ, bhh + (size_t)l * GSZ,
            C + (size_t)l * HSZ, xpad,
            out + 1000 + (size_t)l * HSZ,   // hs[l]
            out + 5000 + (size_t)l * HSZ,   // cs[l]
            out,                            // final output
            (l == LAY - 1) ? 1 : 0);
    }
}